// CMF_76072460747208
// MI455X (gfx1250) — compile-verified
//
#include <hip/hip_runtime.h>
#include <hip/hip_bf16.h>

typedef __attribute__((ext_vector_type(2))) float v2f;
typedef __attribute__((ext_vector_type(8))) float v8f;

#define D128 128
#define NTOK 37
#define BATCH 1024
#define LDSTRIDE 68   // 64 + 4 pad words -> conflict-free 16-row b64 fragment loads

// ---------------------------------------------------------------------------
// Wave-level layer norm helper (wave32, lane holds 4 channels)
// ---------------------------------------------------------------------------
__device__ __forceinline__ float4 wave_ln(float4 x, float4 g, float4 b) {
  float s  = x.x + x.y + x.z + x.w;
  float sq = x.x*x.x + x.y*x.y + x.z*x.z + x.w*x.w;
#pragma unroll
  for (int off = 16; off >= 1; off >>= 1) {
    s  += __shfl_xor(s,  off, 32);
    sq += __shfl_xor(sq, off, 32);
  }
  float m   = s * (1.0f / 128.0f);
  float var = sq * (1.0f / 128.0f) - m * m;
  float inv = rsqrtf(var + 1e-6f);
  float4 r;
  r.x = (x.x - m) * inv * g.x + b.x;
  r.y = (x.y - m) * inv * g.y + b.y;
  r.z = (x.z - m) * inv * g.z + b.z;
  r.w = (x.w - m) * inv * g.w + b.w;
  return r;
}

// ---------------------------------------------------------------------------
// Shared GEMM core: block = 256 thr = 8 waves, block tile M=64 x N=128.
// Wave w: rows [ (w&1)*32, +32 ), cols [ (w>>1)*32, +32 )  (2x2 WMMA tiles).
// Inner step: 4 ds_load_b64 fragments feed 4 independent v_wmma chains.
// ---------------------------------------------------------------------------
struct WaveTile {
  v8f c00, c01, c10, c11;
};

__device__ __forceinline__ void gemm_step(const float* __restrict__ Alds,
                                          const float* __restrict__ Blds,
                                          int rbase, int nbase, int arow, int sel,
                                          WaveTile& t) {
  const float* Ab0 = &Alds[(rbase + arow) * LDSTRIDE + sel * 2];
  const float* Ab1 = Ab0 + 16 * LDSTRIDE;
  const float* Bb0 = &Blds[(nbase + arow) * LDSTRIDE + sel * 2];
  const float* Bb1 = Bb0 + 16 * LDSTRIDE;
#pragma unroll
  for (int kk = 0; kk < 16; ++kk) {
    v2f a0 = *(const v2f*)(Ab0 + kk * 4);
    v2f a1 = *(const v2f*)(Ab1 + kk * 4);
    v2f b0 = *(const v2f*)(Bb0 + kk * 4);
    v2f b1 = *(const v2f*)(Bb1 + kk * 4);
    t.c00 = __builtin_amdgcn_wmma_f32_16x16x4_f32(false, a0, false, b0, (short)0, t.c00, false, false);
    t.c01 = __builtin_amdgcn_wmma_f32_16x16x4_f32(false, a0, false, b1, (short)0, t.c01, false, false);
    t.c10 = __builtin_amdgcn_wmma_f32_16x16x4_f32(false, a1, false, b0, (short)0, t.c10, false, false);
    t.c11 = __builtin_amdgcn_wmma_f32_16x16x4_f32(false, a1, false, b1, (short)0, t.c11, false, false);
  }
}

// ---------------------------------------------------------------------------
// Conv patch-embed as implicit-im2col GEMM with f32 WMMA.
// Out rows r = b*36 + p (36 patches, 6x6), cols = 128 output channels.
// Writes tokens[b][1+p][col] = conv + bias + pos  (pre-LN).
// ---------------------------------------------------------------------------
template <int PS, int PAD, int HW>
__global__ __launch_bounds__(256) void conv_gemm(
    const float* __restrict__ x,     // (B,128,HW,HW)
    const float* __restrict__ W,     // (128, 128*PS*PS) == OIHW flattened
    const float* __restrict__ bias,  // (128)
    const float* __restrict__ pos,   // (36,128)
    float* __restrict__ outTok)      // (B,37,128), rows 1..36
{
  constexpr int K = 128 * PS * PS;
  __shared__ float Alds[64 * LDSTRIDE];
  __shared__ float Blds[128 * LDSTRIDE];

  const int tid  = threadIdx.x;
  const int lane = tid & 31;
  const int wv   = tid >> 5;
  const int m0   = blockIdx.x * 64;
  const int rbase = (wv & 1) * 32;
  const int nbase = (wv >> 1) * 32;
  const int arow = lane & 15;
  const int sel  = lane >> 4;

  // A-panel loader: each thread owns one row (256 % 64 == 0) and 4 k-quads.
  const int lrow = tid & 63;
  const int kqb  = tid >> 6;              // 0..3
  const int rA   = m0 + lrow;
  const int bA   = rA / 36;
  const int pA   = rA - bA * 36;
  const int ph   = pA / 6;
  const int pw   = pA - ph * 6;

  WaveTile t = {};

  for (int k0 = 0; k0 < K; k0 += 64) {
    // stage A: 64x64 im2col gather
#pragma unroll
    for (int qd = 0; qd < 4; ++qd) {
      int kq = kqb + qd * 4;              // 0..15
#pragma unroll
      for (int j = 0; j < 4; ++j) {
        int k   = k0 + kq * 4 + j;
        int cch = k / (PS * PS);
        int rem = k - cch * (PS * PS);
        int kh  = rem / PS;
        int kw  = rem - kh * PS;
        int ih  = ph * PS - PAD + kh;
        int iw  = pw * PS - PAD + kw;
        float v = 0.0f;
        if ((unsigned)ih < (unsigned)HW && (unsigned)iw < (unsigned)HW)
          v = x[((bA * 128 + cch) * HW + ih) * HW + iw];
        Alds[lrow * LDSTRIDE + kq * 4 + j] = v;
      }
    }
    // stage B: 128x64 slice of W (n-major)
#pragma unroll
    for (int i = 0; i < 8; ++i) {
      int idx = tid + i * 256;             // 0..2047 float4s
      int n   = idx >> 4;
      int k4  = idx & 15;
      *(float4*)&Blds[n * LDSTRIDE + k4 * 4] =
          *(const float4*)&W[n * K + k0 + k4 * 4];
    }
    __syncthreads();
    gemm_step(Alds, Blds, rbase, nbase, arow, sel, t);
    __syncthreads();
  }

  const int col0 = nbase + arow;
  const int col1 = col0 + 16;
#pragma unroll
  for (int j = 0; j < 8; ++j) {
    int rl = rbase + j + sel * 8;
    {
      int r = m0 + rl;
      int b = r / 36, p = r - b * 36;
      int o = (b * NTOK + 1 + p) * D128;
      outTok[o + col0] = t.c00[j] + bias[col0] + pos[p * D128 + col0];
      outTok[o + col1] = t.c01[j] + bias[col1] + pos[p * D128 + col1];
    }
    {
      int r = m0 + rl + 16;
      int b = r / 36, p = r - b * 36;
      int o = (b * NTOK + 1 + p) * D128;
      outTok[o + col0] = t.c10[j] + bias[col0] + pos[p * D128 + col0];
      outTok[o + col1] = t.c11[j] + bias[col1] + pos[p * D128 + col1];
    }
  }
}

// ---------------------------------------------------------------------------
// Linear GEMM: out[M x 128] = A[M x 128] * W^T (+ bias + resid).  K = 128.
// ---------------------------------------------------------------------------
__global__ __launch_bounds__(256) void linear_gemm(
    const float* __restrict__ A,
    const float* __restrict__ W,      // (128,128), row n dotted with A row
    float* __restrict__ out,
    const float* __restrict__ bias,   // nullable (paired with resid)
    const float* __restrict__ resid)  // nullable
{
  __shared__ float Alds[64 * LDSTRIDE];
  __shared__ float Blds[128 * LDSTRIDE];

  const int tid  = threadIdx.x;
  const int lane = tid & 31;
  const int wv   = tid >> 5;
  const int m0   = blockIdx.x * 64;
  const int rbase = (wv & 1) * 32;
  const int nbase = (wv >> 1) * 32;
  const int arow = lane & 15;
  const int sel  = lane >> 4;

  WaveTile t = {};

  for (int k0 = 0; k0 < 128; k0 += 64) {
    // A: 64x64 -> 1024 float4, 4 per thread
#pragma unroll
    for (int i = 0; i < 4; ++i) {
      int idx = tid + i * 256;
      int row = idx >> 4;
      int k4  = idx & 15;
      *(float4*)&Alds[row * LDSTRIDE + k4 * 4] =
          *(const float4*)&A[(m0 + row) * D128 + k0 + k4 * 4];
    }
    // B: 128x64
#pragma unroll
    for (int i = 0; i < 8; ++i) {
      int idx = tid + i * 256;
      int n   = idx >> 4;
      int k4  = idx & 15;
      *(float4*)&Blds[n * LDSTRIDE + k4 * 4] =
          *(const float4*)&W[n * 128 + k0 + k4 * 4];
    }
    __syncthreads();
    gemm_step(Alds, Blds, rbase, nbase, arow, sel, t);
    __syncthreads();
  }

  const int col0 = nbase + arow;
  const int col1 = col0 + 16;
#pragma unroll
  for (int j = 0; j < 8; ++j) {
    int r0 = m0 + rbase + j + sel * 8;
    int r1 = r0 + 16;
    float v00 = t.c00[j], v01 = t.c01[j], v10 = t.c10[j], v11 = t.c11[j];
    if (bias) {
      v00 += bias[col0] + resid[r0 * D128 + col0];
      v01 += bias[col1] + resid[r0 * D128 + col1];
      v10 += bias[col0] + resid[r1 * D128 + col0];
      v11 += bias[col1] + resid[r1 * D128 + col1];
    }
    out[r0 * D128 + col0] = v00;
    out[r0 * D128 + col1] = v01;
    out[r1 * D128 + col0] = v10;
    out[r1 * D128 + col1] = v11;
  }
}

// ---------------------------------------------------------------------------
// Cls insertion + embed-LN (in place) + block-LN (to lnOut).
// One wave per token; lane holds 4 channels.
// ---------------------------------------------------------------------------
__global__ __launch_bounds__(256) void finalize_embed(
    float* __restrict__ emb,              // (B,37,128): rows 1..36 pre-LN
    const float* __restrict__ clsParam,   // (128) used when prevOut == null
    const float* __restrict__ prevOut,    // (B,37,128): row 0 = class token
    const float* __restrict__ g, const float* __restrict__ be,
    const float* __restrict__ g2, const float* __restrict__ b2,
    float* __restrict__ lnOut)            // (B,37,128)
{
  const int wv   = threadIdx.x >> 5;
  const int lane = threadIdx.x & 31;
  const int token = blockIdx.x * 8 + wv;
  const int b = token / NTOK;
  const int n = token - b * NTOK;
  const int d = lane * 4;

  float4 x;
  if (n == 0) {
    const float* src = prevOut ? (prevOut + (size_t)b * NTOK * D128) : clsParam;
    x = *(const float4*)(src + d);
  } else {
    x = *(const float4*)(emb + (size_t)token * D128 + d);
  }

  float4 gv  = *(const float4*)(g + d);
  float4 bv  = *(const float4*)(be + d);
  float4 y   = wave_ln(x, gv, bv);
  *(float4*)(emb + (size_t)token * D128 + d) = y;

  float4 g2v = *(const float4*)(g2 + d);
  float4 b2v = *(const float4*)(b2 + d);
  float4 z   = wave_ln(y, g2v, b2v);
  *(float4*)(lnOut + (size_t)token * D128 + d) = z;
}

// ---------------------------------------------------------------------------
// Cross attention core: block = 1 batch, wave = 1 head (hd=32, N=37).
// Row strides padded to 33 words -> conflict-free per-lane row access.
// ---------------------------------------------------------------------------
__global__ __launch_bounds__(128) void attn_kernel(
    const float* __restrict__ q, const float* __restrict__ k,
    const float* __restrict__ v, float* __restrict__ o)
{
  __shared__ float qs[4][NTOK][33];
  __shared__ float ks[4][NTOK][33];
  __shared__ float vs[4][NTOK][33];
  __shared__ float pr[4][40];

  const int h    = threadIdx.x >> 5;
  const int lane = threadIdx.x & 31;
  const int b    = blockIdx.x;
  const int base = (b * NTOK) * D128 + h * 32 + lane;

  for (int m = 0; m < NTOK; ++m) {
    qs[h][m][lane] = q[base + m * D128];
    ks[h][m][lane] = k[base + m * D128];
    vs[h][m][lane] = v[base + m * D128];
  }

  const float scale = 0.17677669529663687f;  // 32^-0.5
  const bool  has2  = lane < 5;
  const int   m2    = has2 ? (32 + lane) : 36;

  for (int n = 0; n < NTOK; ++n) {
    float s1 = 0.0f, s2 = 0.0f;
#pragma unroll
    for (int d = 0; d < 32; ++d) {
      float qd = qs[h][n][d];
      s1 += qd * ks[h][lane][d];
      s2 += qd * ks[h][m2][d];
    }
    s1 *= scale;
    s2 *= scale;
    float mx = fmaxf(s1, has2 ? s2 : -1e30f);
#pragma unroll
    for (int off = 16; off >= 1; off >>= 1) mx = fmaxf(mx, __shfl_xor(mx, off, 32));
    float e1 = __expf(s1 - mx);
    float e2 = has2 ? __expf(s2 - mx) : 0.0f;
    float sum = e1 + e2;
#pragma unroll
    for (int off = 16; off >= 1; off >>= 1) sum += __shfl_xor(sum, off, 32);
    float inv = 1.0f / sum;
    pr[h][lane] = e1 * inv;
    if (has2) pr[h][32 + lane] = e2 * inv;

    float accv = 0.0f;
#pragma unroll
    for (int m = 0; m < NTOK; ++m) accv += pr[h][m] * vs[h][m][lane];
    o[base + n * D128] = accv;
  }
}

// ---------------------------------------------------------------------------
// Host orchestration
// ---------------------------------------------------------------------------
extern "C" void kernel_launch(void* const* d_in, const int* in_sizes, int n_in,
                              void* d_out, int out_size, void* d_ws, size_t ws_size,
                              hipStream_t stream) {
  (void)in_sizes; (void)n_in; (void)out_size; (void)ws_size;

  // setup_inputs() dict order: x1_C,x2_C,x3_C,x1_U,x2_U,x3_U, then params
  // flattened in insertion order: e1C,e1U,e2C,e2U,e3C,e3U (w,b,pos,cls,g,be)
  // then A1,A2,A3 (g1,b1,g2,b2,wq,wk,wv,wo,bo).
  const float* xin[3][2] = {
      {(const float*)d_in[0], (const float*)d_in[3]},
      {(const float*)d_in[1], (const float*)d_in[4]},
      {(const float*)d_in[2], (const float*)d_in[5]}};

  float* out = (float*)d_out;
  float* ws  = (float*)d_ws;
  const size_t TOK = (size_t)BATCH * NTOK * D128;  // 4,849,664 floats
  float* embA = ws + 0 * TOK;
  float* embB = ws + 1 * TOK;
  float* lnA  = ws + 2 * TOK;
  float* lnB  = ws + 3 * TOK;
  float* qb   = ws + 4 * TOK;
  float* kb   = ws + 5 * TOK;
  float* vb   = ws + 6 * TOK;
  float* ob   = ws + 7 * TOK;

  const int convGrid = (BATCH * 36) / 64;   // 576
  const int linGrid  = (BATCH * NTOK) / 64; // 592
  const int finGrid  = (BATCH * NTOK) / 8;  // 4736

  for (int s = 0; s < 3; ++s) {
    const int eC = 6 + s * 12;
    const int eU = eC + 6;
    const int af = 42 + s * 9;
    auto P = [&](int i) { return (const float*)d_in[i]; };

    if (s == 0) {
      conv_gemm<3, 0, 19><<<convGrid, 256, 0, stream>>>(
          xin[s][0], P(eC + 0), P(eC + 1), P(eC + 2), embA);
      conv_gemm<3, 0, 19><<<convGrid, 256, 0, stream>>>(
          xin[s][1], P(eU + 0), P(eU + 1), P(eU + 2), embB);
    } else {
      conv_gemm<2, 1, 10><<<convGrid, 256, 0, stream>>>(
          xin[s][0], P(eC + 0), P(eC + 1), P(eC + 2), embA);
      conv_gemm<2, 1, 10><<<convGrid, 256, 0, stream>>>(
          xin[s][1], P(eU + 0), P(eU + 1), P(eU + 2), embB);
    }

    const float* prev = (s == 0) ? nullptr : (const float*)out;
    finalize_embed<<<finGrid, 256, 0, stream>>>(
        embA, P(eC + 3), prev, P(eC + 4), P(eC + 5), P(af + 0), P(af + 1), lnA);
    finalize_embed<<<finGrid, 256, 0, stream>>>(
        embB, P(eU + 3), prev, P(eU + 4), P(eU + 5), P(af + 2), P(af + 3), lnB);

    linear_gemm<<<linGrid, 256, 0, stream>>>(lnA, P(af + 4), qb, nullptr, nullptr);
    linear_gemm<<<linGrid, 256, 0, stream>>>(lnB, P(af + 5), kb, nullptr, nullptr);
    linear_gemm<<<linGrid, 256, 0, stream>>>(lnB, P(af + 6), vb, nullptr, nullptr);

    attn_kernel<<<BATCH, 128, 0, stream>>>(qb, kb, vb, ob);

    // out = o * wo^T + bo + embB ; d_out doubles as t-source for next stage
    linear_gemm<<<linGrid, 256, 0, stream>>>(ob, P(af + 7), out, P(af + 8), embB);
  }
}